// FreqSetDPPAttention_23897198035719
// MI455X (gfx1250) — compile-verified
//
#include <hip/hip_runtime.h>
#include <math.h>

#define NN 16
#define VV 207
#define LIN 12
#define DD 64
#define NC 12
#define NEGV (-1e20f)

typedef __attribute__((ext_vector_type(2))) float v2f;
typedef __attribute__((ext_vector_type(8))) float v8f;
typedef __attribute__((ext_vector_type(4))) unsigned int v4u;
typedef __attribute__((ext_vector_type(8))) int v8i;
typedef __attribute__((ext_vector_type(4))) int v4i;

#if defined(__has_builtin)
#if __has_builtin(__builtin_amdgcn_tensor_load_to_lds) && __has_builtin(__builtin_amdgcn_s_wait_tensorcnt)
#define HAVE_TDM 1
#endif
#endif
#ifndef HAVE_TDM
#define HAVE_TDM 0
#endif

// workspace layout (floats)
#define OFF_XM 0
#define OFF_Q  (OFF_XM + NN*NC*DD)
#define OFF_KF (OFF_Q  + NN*NC*DD)
#define OFF_KT (OFF_KF + NN*NC*DD*DD)
#define OFF_B  (OFF_KT + NN*NC*DD)
#define OFF_R  (OFF_B  + NN*NC*NC*DD)
#define OFF_S  (OFF_R  + NN*NC*NC)
#define OFF_W  (OFF_S  + NN*NC*NC*NC)

// ---------------- x_mean[n,i,c] = mean over (v,l_in) ----------------
__global__ void k_xmean(const float* __restrict__ x, float* __restrict__ xm) {
  int blk = blockIdx.x;              // n*NC + i
  int n = blk / NC, i = blk % NC;
  int c = threadIdx.x;               // 64 threads
  const float* p = x + (size_t)n*VV*LIN*NC*DD + (size_t)i*DD + c;
  float s = 0.f;
  for (int vl = 0; vl < VV*LIN; ++vl) s += p[(size_t)vl*NC*DD];
  xm[blk*DD + c] = s * (1.0f/(float)(VV*LIN));
}

// ---------------- kfull[n,i,:,:] = xm[n,i,:] @ wk ; kt = mean_d ----------------
__global__ void k_kfull(const float* __restrict__ xm, const float* __restrict__ wk,
                        float* __restrict__ kf, float* __restrict__ kt) {
  __shared__ float sx[DD];
  __shared__ float skf[DD*DD];
  int blk = blockIdx.x;              // n*NC + i
  int tid = threadIdx.x;             // 256
  if (tid < DD) sx[tid] = xm[blk*DD + tid];
  __syncthreads();
  for (int j = tid; j < DD*DD; j += 256) {
    float s = 0.f;
    for (int m = 0; m < DD; ++m) s += sx[m] * wk[m*DD*DD + j];
    skf[j] = s;
    kf[(size_t)blk*DD*DD + j] = s;
  }
  __syncthreads();
  if (tid < DD) {
    float s = 0.f;
    for (int d = 0; d < DD; ++d) s += skf[tid*DD + d];
    kt[blk*DD + tid] = s * (1.0f/(float)DD);
  }
}

// ---------------- q[n,:,:] = xm[n] @ wq ----------------
__global__ void k_q(const float* __restrict__ xm, const float* __restrict__ wq,
                    float* __restrict__ q) {
  __shared__ float sx[NC*DD];
  int n = blockIdx.x; int tid = threadIdx.x; // 256
  for (int j = tid; j < NC*DD; j += 256) sx[j] = xm[n*NC*DD + j];
  __syncthreads();
  for (int o = tid; o < NC*DD; o += 256) {
    int i = o / DD, d = o % DD;
    float s = 0.f;
    for (int m = 0; m < DD; ++m) s += sx[i*DD + m] * wq[m*DD + d];
    q[n*NC*DD + o] = s;
  }
}

// ---------------- b[n,q,i,c] = q[n,q,:] . kfull[n,i,c,:] ; r = ||b||_c ----------------
__global__ void k_bmat(const float* __restrict__ q, const float* __restrict__ kf,
                       float* __restrict__ b, float* __restrict__ r) {
  __shared__ float red[DD];
  int blk = blockIdx.x;              // (n*NC+qi)*NC + i
  int i = blk % NC; int nq = blk / NC; int n = nq / NC;
  int c = threadIdx.x;               // 64
  const float* qp = q + (size_t)nq*DD;
  const float* kp = kf + ((size_t)(n*NC + i))*DD*DD + (size_t)c*DD;
  float s = 0.f;
  for (int d = 0; d < DD; ++d) s += qp[d] * kp[d];
  b[(size_t)blk*DD + c] = s;
  red[c] = s*s;
  __syncthreads();
  if (c == 0) {
    float t = 0.f;
    for (int k = 0; k < DD; ++k) t += red[k];
    r[blk] = sqrtf(t);
  }
}

// ---------------- s[n,q,i,j] = (f_i . f_j + 1)/2, f = b/r ----------------
__global__ void k_smat(const float* __restrict__ b, const float* __restrict__ r,
                       float* __restrict__ s) {
  __shared__ float sb[NC*DD];
  __shared__ float sr[NC];
  int nq = blockIdx.x;               // n*NC + qi
  int tid = threadIdx.x;             // 144
  for (int j = tid; j < NC*DD; j += 144) sb[j] = b[(size_t)nq*NC*DD + j];
  if (tid < NC) sr[tid] = r[nq*NC + tid];
  __syncthreads();
  int i = tid / NC, j = tid % NC;
  float dot = 0.f;
  for (int c = 0; c < DD; ++c) dot += sb[i*DD + c] * sb[j*DD + c];
  dot /= (sr[i]*sr[j]);
  s[(size_t)nq*NC*NC + tid] = (dot + 1.0f)*0.5f;
}

// ---------------- greedy DPP MAP + masked softmax -> attn weights W[n,q,i] ----------------
__global__ void k_dpp(const float* __restrict__ s, const float* __restrict__ r,
                      const float* __restrict__ q, const float* __restrict__ kt,
                      float* __restrict__ W) {
  int p = blockIdx.x*32 + threadIdx.x;   // 0..191 problems
  if (p >= NN*NC) return;
  int n = p / NC, node = p % NC;
  float l[NC*NC], rs[NC], diag[NC], cc[NC*NC], ev[NC];
  bool sel[NC];
  for (int k = 0; k < NC*NC; ++k) l[k] = s[(size_t)p*NC*NC + k];
  for (int k = 0; k < NC; ++k) rs[k] = r[p*NC + k];
  for (int k = 0; k < NC; ++k) { diag[k] = l[k*NC + k]; sel[k] = false; }
  diag[node] = NEGV;
  for (int k = 0; k < NC; ++k) { l[node*NC + k] = 0.f; l[k*NC + node] = 0.f; }
  int j = 0; float best = -INFINITY;
  for (int k = 0; k < NC; ++k) { float v = diag[k]*rs[k]; if (v > best) { best = v; j = k; } }
  sel[j] = true;
  for (int k = 0; k < NC*NC; ++k) cc[k] = 0.f;
  int it = 1; bool done = false;
  while (it < NC && !done) {
    float inv = 1.0f / sqrtf(diag[j]);
    for (int i = 0; i < NC; ++i) {
      bool m = (!sel[i]) && (i != node);
      float d = 0.f;
      for (int rr2 = 0; rr2 < it; ++rr2) d += cc[rr2*NC + j] * cc[rr2*NC + i];
      float e = (l[j*NC + i] - d) * inv;
      ev[i] = m ? e : 0.f;
    }
    for (int i = 0; i < NC; ++i) { cc[it*NC + i] = ev[i]; diag[i] -= ev[i]*ev[i]; }
    diag[j] = NEGV;
    int j2 = 0; best = -INFINITY;
    for (int k = 0; k < NC; ++k) { float v = diag[k]*rs[k]; if (v > best) { best = v; j2 = k; } }
    bool stop = diag[j2]*rs[j2] < 1.0f;
    if (!stop) sel[j2] = true;
    j = j2; ++it; done = stop;
  }
  const float* qp = q + (size_t)p*DD;
  float logit[NC]; float mx = -INFINITY;
  for (int i = 0; i < NC; ++i) {
    if (sel[i]) {
      float d = 0.f;
      const float* kp = kt + (size_t)(n*NC + i)*DD;
      for (int c = 0; c < DD; ++c) d += qp[c]*kp[c];
      logit[i] = d; if (d > mx) mx = d;
    } else logit[i] = 0.f;
  }
  float ssum = 0.f;
  for (int i = 0; i < NC; ++i) {
    float e = sel[i] ? expf(logit[i]-mx) : 0.f;
    logit[i] = e; ssum += e;
  }
  for (int i = 0; i < NC; ++i) W[p*NC + i] = logit[i] / ssum;
}

// ---------------- fused main pass helpers ----------------
#define AST 68   // padded LDS row stride (floats): bank-conflict-free for WMMA gathers

// Stage `rows` rows of 64 f32 from gsrc into LDS (row stride AST floats).
// TDM path: each wave DMAs its quarter with hardware pad (4 DW after every 64 DW),
// which reproduces the stride-68 layout with zero VGPR traffic (TENSORcnt tracked).
__device__ __forceinline__ void stage_rows(const float* __restrict__ gsrc, float* ldst,
                                           int rows, int tid) {
#if HAVE_TDM
  int wave = tid >> 5;
  int wrows = rows >> 2;                         // rows per wave (12 or 16)
  const float* gp = gsrc + (size_t)(wave * wrows) * 64;
  unsigned long long ga = (unsigned long long)(uintptr_t)gp;
  unsigned int ga_lo = __builtin_amdgcn_readfirstlane((unsigned int)ga);
  unsigned int ga_hi = __builtin_amdgcn_readfirstlane((unsigned int)(ga >> 32));
  unsigned int lds   = __builtin_amdgcn_readfirstlane(
      (unsigned int)(uintptr_t)(ldst + wave * wrows * AST));
  unsigned int nrows = __builtin_amdgcn_readfirstlane((unsigned int)wrows);
  v4u g0;
  g0.x = 1u;                                     // count=1 (valid), load, no gather
  g0.y = lds;                                    // D# lds_addr (bytes)
  g0.z = ga_lo;                                  // global_addr[31:0]
  g0.w = (ga_hi & 0x01FFFFFFu) | 0x80000000u;    // global_addr[56:32] | type=2
  v8i g1;
  g1[0] = (int)((2u << 16)      // data_size = 4B
              | (1u << 20)      // pad_enable
              | (5u << 22)      // pad_interval: 64 DWORDs
              | (3u << 25));    // pad_amount: 4 DWORDs  -> LDS row stride 68
  g1[1] = (int)(64u << 16);                      // tensor_dim0 = 64 (low16)
  g1[2] = (int)(nrows << 16);                    // tensor_dim0 hi=0 | tensor_dim1 lo16
  g1[3] = (int)(64u << 16);                      // tensor_dim1 hi=0 | tile_dim0 = 64
  g1[4] = (int)(nrows & 0xFFFFu);                // tile_dim1 = rows, tile_dim2 = 0
  g1[5] = 64;                                    // tensor_dim0_stride = 64 (low32)
  g1[6] = 0;                                     // stride0 hi | stride1 lo
  g1[7] = 0;
  v4i z = {0, 0, 0, 0};
#if __clang_major__ >= 23
  v8i z8 = {0, 0, 0, 0, 0, 0, 0, 0};
  __builtin_amdgcn_tensor_load_to_lds(g0, g1, z, z, z8, 0);
#else
  __builtin_amdgcn_tensor_load_to_lds(g0, g1, z, z, 0);
#endif
#else
  for (int f4 = tid; f4 < rows*16; f4 += 128) {
    int row = f4 >> 4, cg = f4 & 15;
    *(float4*)&ldst[row*AST + cg*4] = *(const float4*)&gsrc[row*64 + cg*4];
  }
#endif
}

__device__ __forceinline__ void wait_tdm() {
#if HAVE_TDM
  __builtin_amdgcn_s_wait_tensorcnt(0);
#endif
}

// C(48x64) = A(48x64) @ B(64x64) via V_WMMA_F32_16X16X4_F32, 4 waves x 3 tiles
__device__ __forceinline__ void gemm48(const float* sA_, const float* sB_, float* sC_, int tid) {
  int wave = tid >> 5, lane = tid & 31, half = lane >> 4, lid = lane & 15;
  for (int t = wave; t < 12; t += 4) {
    int mt = t >> 2, nt = t & 3;
    v8f acc = {};
    for (int kk = 0; kk < 16; ++kk) {
      int k0 = kk*4 + half*2;
      v2f a, b;
      a.x = sA_[(mt*16 + lid)*AST + k0];
      a.y = sA_[(mt*16 + lid)*AST + k0 + 1];
      b.x = sB_[k0*AST + nt*16 + lid];
      b.y = sB_[(k0 + 1)*AST + nt*16 + lid];
      acc = __builtin_amdgcn_wmma_f32_16x16x4_f32(false, a, false, b, (short)0, acc, false, false);
    }
    for (int rr = 0; rr < 8; ++rr)
      sC_[(mt*16 + half*8 + rr)*AST + nt*16 + lid] = acc[rr];
  }
}

// ---------------- fused: mix -> @wv -> +res,LN -> @w1,gelu -> @w2 -> +res,LN ----------------
__global__ __launch_bounds__(128) void k_fused(
    const float* __restrict__ x, const float* __restrict__ wv,
    const float* __restrict__ ln_w, const float* __restrict__ ln_b,
    const float* __restrict__ w1, const float* __restrict__ b1,
    const float* __restrict__ w2, const float* __restrict__ b2,
    const float* __restrict__ W, float* __restrict__ out) {
  __shared__ __align__(16) float sA[48*AST];   // x block (residual), later gelu(h)
  __shared__ __align__(16) float sM[48*AST];   // xmix, later res (post-LN1)
  __shared__ __align__(16) float sC[48*AST];   // GEMM output scratch
  __shared__ __align__(16) float sB[64*AST];   // staged weight matrix
  __shared__ float sW[NC*NC];
  __shared__ float sLw[DD], sLb[DD], sB1[DD], sB2[DD];
  __shared__ float rMean[48], rRstd[48];

  const int tid = threadIdx.x;
  int g = blockIdx.x;
  int n = g / (VV*3); int rem = g % (VV*3); int v = rem / 3; int li0 = (rem % 3)*4;
  size_t base = ((size_t)(n*VV + v)*LIN + li0) * (size_t)(NC*DD);

  // TDM: 48x64 contiguous x block -> sA (stride 68 via pad), wv -> sB
  stage_rows(x + base, sA, 48, tid);
  stage_rows(wv, sB, 64, tid);
  for (int k = tid; k < NC*NC; k += 128) sW[k] = W[n*NC*NC + k];
  if (tid < DD) { sLw[tid] = ln_w[tid]; sLb[tid] = ln_b[tid]; }
  else          { int t2 = tid - DD; sB1[t2] = b1[t2]; sB2[t2] = b2[t2]; }
  wait_tdm();
  __syncthreads();

  // mixing: xmix[(lq,qi),c] = sum_i W[qi,i] * x[(lq,i),c]
  for (int o = tid; o < 48*64; o += 128) {
    int row = o >> 6, c = o & 63;
    int lq = row / 12, qi = row % 12;
    const float* wrow = &sW[qi*NC];
    float s = 0.f;
    for (int i = 0; i < NC; ++i) s += wrow[i] * sA[(lq*12 + i)*AST + c];
    sM[row*AST + c] = s;
  }
  __syncthreads();

  gemm48(sM, sB, sC, tid);           // attn_out = xmix @ wv
  __syncthreads();

  stage_rows(w1, sB, 64, tid);       // prefetch w1 while LN1 epilogue runs
  // LN1 stats: val = attn_out + residual
  if (tid < 48) {
    float m = 0.f, sq = 0.f;
    for (int c = 0; c < 64; ++c) { float val = sC[tid*AST + c] + sA[tid*AST + c]; m += val; sq += val*val; }
    m *= (1.0f/64.0f);
    float var = sq*(1.0f/64.0f) - m*m;
    rMean[tid] = m; rRstd[tid] = rsqrtf(var + 1e-5f);
  }
  __syncthreads();
  for (int o = tid; o < 48*64; o += 128) {
    int row = o >> 6, c = o & 63;
    float val = sC[row*AST + c] + sA[row*AST + c];
    sM[row*AST + c] = (val - rMean[row]) * rRstd[row] * sLw[c] + sLb[c];
  }
  wait_tdm();
  __syncthreads();

  gemm48(sM, sB, sC, tid);           // h_raw = res @ w1
  __syncthreads();

  stage_rows(w2, sB, 64, tid);       // prefetch w2 while gelu epilogue runs
  for (int o = tid; o < 48*64; o += 128) {
    int row = o >> 6, c = o & 63;
    float hv = sC[row*AST + c] + sB1[c];
    sA[row*AST + c] = 0.5f * hv * (1.0f + erff(hv * 0.70710678118654752440f));
  }
  wait_tdm();
  __syncthreads();

  gemm48(sA, sB, sC, tid);           // ff_raw = gelu(h) @ w2
  __syncthreads();

  // LN2 stats: val = res + ff_raw + b2
  if (tid < 48) {
    float m = 0.f, sq = 0.f;
    for (int c = 0; c < 64; ++c) { float val = sM[tid*AST + c] + sC[tid*AST + c] + sB2[c]; m += val; sq += val*val; }
    m *= (1.0f/64.0f);
    float var = sq*(1.0f/64.0f) - m*m;
    rMean[tid] = m; rRstd[tid] = rsqrtf(var + 1e-5f);
  }
  __syncthreads();
  for (int o = tid; o < 48*64; o += 128) {
    int row = o >> 6, c = o & 63;
    float val = sM[row*AST + c] + sC[row*AST + c] + sB2[c];
    out[base + o] = (val - rMean[row]) * rRstd[row] * sLw[c] + sLb[c];
  }
}

extern "C" void kernel_launch(void* const* d_in, const int* in_sizes, int n_in,
                              void* d_out, int out_size, void* d_ws, size_t ws_size,
                              hipStream_t stream) {
  const float* x   = (const float*)d_in[0];
  const float* wq  = (const float*)d_in[1];
  const float* wk  = (const float*)d_in[2];
  const float* wv  = (const float*)d_in[3];
  const float* lnw = (const float*)d_in[4];
  const float* lnb = (const float*)d_in[5];
  const float* w1  = (const float*)d_in[6];
  const float* b1  = (const float*)d_in[7];
  const float* w2  = (const float*)d_in[8];
  const float* b2  = (const float*)d_in[9];
  float* out = (float*)d_out;
  float* ws  = (float*)d_ws;
  float* xm = ws + OFF_XM; float* q = ws + OFF_Q; float* kf = ws + OFF_KF;
  float* kt = ws + OFF_KT; float* b = ws + OFF_B; float* r  = ws + OFF_R;
  float* s  = ws + OFF_S;  float* W = ws + OFF_W;

  k_xmean<<<NN*NC, DD, 0, stream>>>(x, xm);
  k_kfull<<<NN*NC, 256, 0, stream>>>(xm, wk, kf, kt);
  k_q<<<NN, 256, 0, stream>>>(xm, wq, q);
  k_bmat<<<NN*NC*NC, DD, 0, stream>>>(q, kf, b, r);
  k_smat<<<NN*NC, 144, 0, stream>>>(b, r, s);
  k_dpp<<<6, 32, 0, stream>>>(s, r, q, kt, W);
  k_fused<<<NN*VV*3, 128, 0, stream>>>(x, wv, lnw, lnb, w1, b1, w2, b2, W, out);
}